// InverseCovMatrixFromPositions_26757646254417
// MI455X (gfx1250) — compile-verified
//
#include <hip/hip_runtime.h>
#include <hip/hip_bf16.h>

// Problem constants (match reference)
#define SIGMA_SQ 1.0f
#define PHI      0.5f
#define TAU      0.01f
#define NMAT     20     // neighbor size -> 20x20 matrix
#define NEDGE    10
#define WAVES_PER_BLOCK 8
#define ROWSTRIDE 21    // LDS row stride in floats (odd -> conflict-free column access)

typedef __attribute__((ext_vector_type(2))) float v2f;
typedef __attribute__((ext_vector_type(8))) float v8f;

__device__ __forceinline__ float readlane_f(float v, int srclane) {
    return __uint_as_float(
        (unsigned)__builtin_amdgcn_readlane((int)__float_as_uint(v), srclane));
}

// Process one 16x16 WMMA Gram tile: fuse distance -> exp covariance -> nugget
// -> edge mask, scatter valid elements of cov_final into the wave's LDS matrix.
template <int R, int C, int RMAX>
__device__ __forceinline__ void tile_to_cov(
    const v8f& c, const float* qs, float qn, unsigned selmask,
    int lane, int hi, int l15, float* smrow /* smat[wave] */)
{
    const int nn = C * 16 + l15;                       // this lane's column
    const unsigned seln = (selmask >> nn) & 1u;
    #pragma unroll
    for (int r = 0; r < RMAX; ++r) {
        // row index of c[r]; for R==1 only hi==0 rows (16..19) survive the guard
        const int m = R * 16 + hi * 8 + r;
        float qm;
        if (R == 0) qm = hi ? qs[8 + r] : qs[r];       // two SGPR refs + cndmask
        else        qm = qs[16 + r];                   // hi==1 lanes discarded below
        float d2   = qm + qn - 2.f * c[r];
        float dist = __builtin_amdgcn_sqrtf(fmaxf(d2, 0.f));   // raw v_sqrt_f32
        float diag = (m == nn) ? 1.f : 0.f;            // SIGMA_SQ * eye
        float cov  = SIGMA_SQ * __expf(-PHI * dist) + (TAU * SIGMA_SQ) * diag;
        const unsigned selm = (selmask >> m) & 1u;
        float val = (selm & seln) ? cov : diag;
        if (m < NMAT && nn < NMAT)
            smrow[m * ROWSTRIDE + nn] = val;
    }
}

__global__ __launch_bounds__(WAVES_PER_BLOCK * 32)
void inv_cov_gj_wmma_kernel(const float* __restrict__ pos,    // [B, 40]
                            const int*   __restrict__ edges,  // [B, 10]
                            float*       __restrict__ out,    // [B, 20, 20]
                            int B)
{
    __shared__ float smat[WAVES_PER_BLOCK][NMAT * ROWSTRIDE];

    const int lane = threadIdx.x & 31;
    const int wave = threadIdx.x >> 5;
    const int b    = blockIdx.x * WAVES_PER_BLOCK + wave;
    if (b >= B) return;                      // wave-uniform

    const int hi  = lane >> 4;               // 0: lanes 0-15, 1: lanes 16-31
    const int l15 = lane & 15;

    // ---- load this lane's point (lane i owns point i; lanes >=20 own zeros)
    float px = 0.f, py = 0.f, q = 0.f;
    if (lane < NMAT) {
        px = pos[(size_t)b * 40 + 2 * lane];
        py = pos[(size_t)b * 40 + 2 * lane + 1];
        q  = px * px + py * py;
    }

    // ---- selection bitmask from edge_list: OR-reduce, then force scalar
    unsigned selv = 0;
    if (lane < NEDGE) {
        int e = edges[(size_t)b * NEDGE + lane];
        selv = 1u << (e & 31);
    }
    #pragma unroll
    for (int off = 16; off > 0; off >>= 1)
        selv |= (unsigned)__shfl_xor((int)selv, off, 32);
    const unsigned selmask = (unsigned)__builtin_amdgcn_readfirstlane((int)selv);

    // ---- broadcast all |p_i|^2 to SGPRs once (static-lane readlane)
    float qs[NMAT];
    #pragma unroll
    for (int i = 0; i < NMAT; ++i) qs[i] = readlane_f(q, i);

    // ---- marshal WMMA operands. A 16x4 f32: lanes 0-15 hold (M=lane,K=0/1),
    // lanes 16-31 hold K=2/3 zero pad. Gram is symmetric -> B operand == A operand.
    float px0  = __shfl(px, l15, 32);
    float py0  = __shfl(py, l15, 32);
    float px16 = __shfl(px, 16 + l15, 32);
    float py16 = __shfl(py, 16 + l15, 32);
    float qn0  = __shfl(q, l15, 32);         // q[col] for C=0 tiles
    float qn16 = __shfl(q, 16 + l15, 32);    // q[col] for C=1 tiles (0 for col>=20)
    v2f A0, A1;
    A0.x = hi ? 0.f : px0;   A0.y = hi ? 0.f : py0;
    A1.x = hi ? 0.f : px16;  A1.y = hi ? 0.f : py16;

    const v8f zc = {0.f, 0.f, 0.f, 0.f, 0.f, 0.f, 0.f, 0.f};
    v8f c00 = __builtin_amdgcn_wmma_f32_16x16x4_f32(false, A0, false, A0, (short)0, zc, false, false);
    v8f c01 = __builtin_amdgcn_wmma_f32_16x16x4_f32(false, A0, false, A1, (short)0, zc, false, false);
    v8f c10 = __builtin_amdgcn_wmma_f32_16x16x4_f32(false, A1, false, A0, (short)0, zc, false, false);
    v8f c11 = __builtin_amdgcn_wmma_f32_16x16x4_f32(false, A1, false, A1, (short)0, zc, false, false);

    float* smrow = smat[wave];
    tile_to_cov<0, 0, 8>(c00, qs, qn0,  selmask, lane, hi, l15, smrow);
    tile_to_cov<0, 1, 8>(c01, qs, qn16, selmask, lane, hi, l15, smrow);
    tile_to_cov<1, 0, 4>(c10, qs, qn0,  selmask, lane, hi, l15, smrow);
    tile_to_cov<1, 1, 4>(c11, qs, qn16, selmask, lane, hi, l15, smrow);

    // same-wave DS ops are in-order; fence stops compiler reordering store->load
    asm volatile("s_wait_dscnt 0x0" ::: "memory");

    // ---- gather rows: lane i owns row i of A (lanes >=20 get zeros)
    float arow2[NMAT];
    #pragma unroll
    for (int j = 0; j < NMAT; ++j)
        arow2[j] = (lane < NMAT) ? smrow[lane * ROWSTRIDE + j] : 0.f;

    float inv[NMAT];
    #pragma unroll
    for (int j = 0; j < NMAT; ++j)
        inv[j] = (lane == j) ? 1.f : 0.f;

    // ---- Gauss-Jordan, fully unrolled; identity pivots skipped by SCALAR branch.
    // Deferred pivot-row scaling: each pivoted row stays unscaled in place
    // (elimination is linear, so stored == true/pr invariantly); the lane just
    // remembers pr and rescales its inverse row once at the end. The broadcast
    // scale folds into the per-lane multiplier: inner op = readlane + fma only.
    float myscale = 1.f;
    #pragma unroll
    for (int k = 0; k < NMAT; ++k) {
        if (selmask & (1u << k)) {
            const float piv = readlane_f(arow2[k], k);
            float pr = __builtin_amdgcn_rcpf(piv);
            pr = fmaf(fmaf(-piv, pr, 1.f), pr, pr);    // one Newton step
            const bool isP = (lane == k);
            const float mp = (isP ? 0.f : arow2[k]) * pr;  // pivot lane: no-op
            myscale = isP ? pr : myscale;
            // A columns j > k (columns j < k are already e_j -> no-op)
            #pragma unroll
            for (int j = k + 1; j < NMAT; ++j)
                arow2[j] = fmaf(-mp, readlane_f(arow2[j], k), arow2[j]);
            // inverse columns j <= k (columns j > k are still e_j -> no-op)
            #pragma unroll
            for (int j = 0; j <= k; ++j)
                inv[j] = fmaf(-mp, readlane_f(inv[j], k), inv[j]);
        }
    }
    #pragma unroll
    for (int j = 0; j < NMAT; ++j)
        inv[j] *= myscale;                  // apply this row's deferred pivot scale

    // ---- write row i of the inverse: 5x b128 stores, 16B-aligned (b*1600 + lane*80)
    if (lane < NMAT) {
        float4* o = (float4*)(out + (size_t)b * (NMAT * NMAT) + lane * NMAT);
        #pragma unroll
        for (int j = 0; j < 5; ++j) {
            float4 v;
            v.x = inv[4 * j + 0];
            v.y = inv[4 * j + 1];
            v.z = inv[4 * j + 2];
            v.w = inv[4 * j + 3];
            o[j] = v;
        }
    }
}

extern "C" void kernel_launch(void* const* d_in, const int* in_sizes, int n_in,
                              void* d_out, int out_size, void* d_ws, size_t ws_size,
                              hipStream_t stream) {
    (void)n_in; (void)out_size; (void)d_ws; (void)ws_size;
    const float* pos   = (const float*)d_in[0];
    const int*   edges = (const int*)d_in[1];
    float*       out   = (float*)d_out;
    const int B = in_sizes[0] / 40;                 // 100000
    const int blocks = (B + WAVES_PER_BLOCK - 1) / WAVES_PER_BLOCK;
    inv_cov_gj_wmma_kernel<<<blocks, WAVES_PER_BLOCK * 32, 0, stream>>>(pos, edges, out, B);
}